// HistogramBinningCalibrationByFeature_45621142618493
// MI455X (gfx1250) — compile-verified
//
#include <hip/hip_runtime.h>

// HistogramBinningCalibrationByFeature — memory-bound gather/stream kernel.
// No matrix contraction exists in this op, so no WMMA path applies; the
// optimization targets are B128 vector memory ops, NT cache hints for the
// single-touch streams (protecting the 1.7MB bin tables in the 192MB L2),
// and wave32 occupancy.

#define NUM_BINS_C 5000
#define NUM_SEGMENTS_C 42
#define BIN_CTR_W_C 0.9995f
#define ONE_MINUS_BIN_CTR_W_C 0.0005f

typedef float v4f __attribute__((ext_vector_type(4)));
typedef int   v4i __attribute__((ext_vector_type(4)));

__device__ __forceinline__ float fast_sigmoid(float x) {
    // 1 / (1 + e^-x) using v_exp_f32 + v_rcp_f32 (memory-bound kernel; ~1ulp ok)
    float e = __expf(-x);
    return __builtin_amdgcn_rcpf(1.0f + e);
}

__global__ __launch_bounds__(256) void hbc_calibrate_kernel(
    const int*   __restrict__ seg_val,    // [N]
    const int*   __restrict__ seg_len,    // [N+1] (sorted)
    const float* __restrict__ logit,      // [N]
    const float* __restrict__ bin_pos,    // [NUM_INTERVAL]
    const float* __restrict__ bin_num,    // [NUM_INTERVAL]
    const float* __restrict__ pos_w,      // [1]
    float*       __restrict__ out_pred,   // [N]
    int*         __restrict__ out_bin,    // [N]
    int n)
{
    const int g    = blockIdx.x * blockDim.x + threadIdx.x;
    const int base = g * 4;
    if (base >= n) return;

    // Uniform scalar: compiler emits s_load for pos_w[0] (constant cache hit).
    const float lw = __logf(pos_w[0]);

    if (base + 4 <= n) {
        // ---- B128 non-temporal streaming loads (touched exactly once) ----
        v4f lg  = __builtin_nontemporal_load((const v4f*)(logit   + base));
        v4i sl  = __builtin_nontemporal_load((const v4i*)(seg_len + base));
        int sl4 = __builtin_nontemporal_load(seg_len + base + 4);

        const float lgl[4] = {lg.x, lg.y, lg.z, lg.w};
        const int   sll[5] = {sl.x, sl.y, sl.z, sl.w, sl4};

        v4f pred;
        v4i bins;
        float pr[4]; int bi[4];

        #pragma unroll
        for (int j = 0; j < 4; ++j) {
            const float p = fast_sigmoid(lgl[j] + lw);

            // seg_len values are always in [0, N): gather unconditionally
            // (matches the reference, which gathers before the where-select).
            // seg_len is sorted -> this gather is quasi-sequential (coalesces).
            const int sv    = seg_val[sll[j]] + 1;
            const bool vld  = (sll[j + 1] > sll[j]) && (sv <= NUM_SEGMENTS_C);
            const int dense = vld ? sv : 0;

            const int bin = ((int)__builtin_ceilf(p * (float)NUM_BINS_C) - 1)
                            + dense * NUM_BINS_C;

            // Bin tables: default RT temporal hint -> stays hot in L2 (1.7MB).
            const float cp = bin_pos[bin];
            const float cn = bin_num[bin];

            const float ctr = (cp * __builtin_amdgcn_rcpf(cn)) * BIN_CTR_W_C
                              + p * ONE_MINUS_BIN_CTR_W_C;
            pr[j] = (cn > 0.0f) ? ctr : p;
            bi[j] = bin;
        }
        pred = (v4f){pr[0], pr[1], pr[2], pr[3]};
        bins = (v4i){bi[0], bi[1], bi[2], bi[3]};

        // ---- B128 non-temporal streaming stores ----
        __builtin_nontemporal_store(pred, (v4f*)(out_pred + base));
        __builtin_nontemporal_store(bins, (v4i*)(out_bin  + base));
    } else {
        // Tail (n not divisible by 4): scalar path.
        for (int i = base; i < n; ++i) {
            const float p = fast_sigmoid(logit[i] + lw);
            const int s0 = seg_len[i];
            const int s1 = seg_len[i + 1];
            const int sv = seg_val[s0] + 1;
            const int dense = ((s1 > s0) && (sv <= NUM_SEGMENTS_C)) ? sv : 0;
            const int bin = ((int)__builtin_ceilf(p * (float)NUM_BINS_C) - 1)
                            + dense * NUM_BINS_C;
            const float cp = bin_pos[bin];
            const float cn = bin_num[bin];
            const float ctr = (cp * __builtin_amdgcn_rcpf(cn)) * BIN_CTR_W_C
                              + p * ONE_MINUS_BIN_CTR_W_C;
            __builtin_nontemporal_store((cn > 0.0f) ? ctr : p, out_pred + i);
            __builtin_nontemporal_store(bin, out_bin + i);
        }
    }
}

extern "C" void kernel_launch(void* const* d_in, const int* in_sizes, int n_in,
                              void* d_out, int out_size, void* d_ws, size_t ws_size,
                              hipStream_t stream) {
    (void)n_in; (void)d_ws; (void)ws_size; (void)out_size;
    const int*   seg_val = (const int*)  d_in[0];  // segment_value   [N]
    const int*   seg_len = (const int*)  d_in[1];  // segment_lengths [N+1]
    const float* logit   = (const float*)d_in[2];  // logit           [N]
    const float* bin_pos = (const float*)d_in[3];  // bin_num_positives
    const float* bin_num = (const float*)d_in[4];  // bin_num_examples
    const float* pos_w   = (const float*)d_in[5];  // positive_weight [1]

    const int n = in_sizes[2];                     // NUM_LOGITS

    float* out_pred = (float*)d_out;               // output 0: calibrated [N] f32
    int*   out_bin  = (int*)d_out + n;             // output 1: bin ids    [N] i32

    const int n4     = (n + 3) / 4;                // 4 elements per lane (B128)
    const int block  = 256;                        // 8 wave32 per block
    const int blocks = (n4 + block - 1) / block;

    hbc_calibrate_kernel<<<blocks, block, 0, stream>>>(
        seg_val, seg_len, logit, bin_pos, bin_num, pos_w, out_pred, out_bin, n);
}